// DWConvBlock3D_16578573763375
// MI455X (gfx1250) — compile-verified
//
#include <hip/hip_runtime.h>
#include <hip/hip_bf16.h>
#include <stdint.h>

// Problem constants (from reference setup_inputs)
#define B_ 2
#define C_ 64
#define D_ 64
#define H_ 128
#define W_ 128
#define S_ (D_*H_*W_)           // 1,048,576 elements per (b,c) slice

// Conv tile: 4 x 8 x 128 outputs per 256-thread block
#define TD 4
#define TH 8
#define LZ (TD+2)               // 6 planes (halo)
#define LY (TH+2)               // 10 rows (halo)
#define LROW 136                // 4 pad + 128 + 4 pad -> interior 16B aligned
#define LDS_ELEMS (LZ*LY*LROW)  // 8160 floats = 32,640 B

#define GAS __attribute__((address_space(1)))
#define LAS __attribute__((address_space(3)))

typedef int v4i __attribute__((vector_size(16)));

#ifdef __has_builtin
#if __has_builtin(__builtin_amdgcn_global_load_async_to_lds_b128)
#define USE_ASYNC_LDS 1
#endif
#if __has_builtin(__builtin_amdgcn_s_wait_asynccnt)
#define HAVE_WAIT_ASYNC 1
#endif
#endif

__device__ __forceinline__ void wait_async_all() {
#if defined(HAVE_WAIT_ASYNC)
    __builtin_amdgcn_s_wait_asynccnt(0);
#elif defined(USE_ASYNC_LDS)
    asm volatile("s_wait_asynccnt 0" ::: "memory");
#endif
}

// ---------------------------------------------------------------------------
// Kernel 0: zero the per-(b,c) sum / sumsq accumulators (2*B*C = 256 floats)
// ---------------------------------------------------------------------------
__global__ void dwcb_init_kernel(float* __restrict__ ws) {
    ws[threadIdx.x] = 0.0f;   // exactly 256 threads, 256 floats
}

// ---------------------------------------------------------------------------
// Kernel 1: depthwise 3x3x3 conv (pad=1) + per-(b,c) sum/sumsq reduction.
// Grid (256 tiles, C, B); block 256 threads (8 waves of 32).
// Input tile staged into LDS with halo via async global->LDS loads.
// ---------------------------------------------------------------------------
__global__ __launch_bounds__(256) void dwcb_conv_stats_kernel(
    const float* __restrict__ x, const float* __restrict__ w,
    float* __restrict__ y, float* __restrict__ ws_sum, float* __restrict__ ws_sumsq)
{
    __shared__ float lds[LDS_ELEMS];
    const int tid = threadIdx.x;
    const int tile = blockIdx.x;        // 0..255  (16 d-tiles x 16 h-tiles)
    const int c = blockIdx.y;
    const int b = blockIdx.z;
    const int d0 = (tile >> 4) * TD;
    const int h0 = (tile & 15) * TH;
    const size_t base = (size_t)(b * C_ + c) * (size_t)S_;

    // 1) zero entire LDS tile (covers all halo regions => implicit zero padding)
    #pragma unroll
    for (int i = 0; i < 32; ++i) {
        int idx = tid + i * 256;
        if (idx < LDS_ELEMS) lds[idx] = 0.0f;
    }
    __syncthreads();

    // 2) stage valid input region: 60 rows x 32 chunks of 16B = 1920 chunks
    for (int cid = tid; cid < LZ * LY * 32; cid += 256) {
        int row = cid >> 5;             // 0..59
        int cx  = (cid & 31) << 2;      // float x index 0..124, step 4
        int pl  = row / LY;
        int yy  = row - pl * LY;
        int gz  = d0 - 1 + pl;
        int gy  = h0 - 1 + yy;
        if ((unsigned)gz < (unsigned)D_ && (unsigned)gy < (unsigned)H_) {
            const float* src = x + base + ((size_t)gz * H_ + gy) * W_ + cx;
            float* dst = &lds[row * LROW + 4 + cx];   // 16B aligned both sides
#ifdef USE_ASYNC_LDS
            __builtin_amdgcn_global_load_async_to_lds_b128(
                (GAS v4i*)src, (LAS v4i*)dst, 0, 0);
#else
            *(float4*)dst = *(const float4*)src;
#endif
        }
    }
    wait_async_all();
    __syncthreads();

    // 3) load 27 taps (block-uniform channel -> scalar loads)
    const float* wp = w + c * 27;
    float wk[27];
    #pragma unroll
    for (int i = 0; i < 27; ++i) wk[i] = wp[i];

    // 4) each thread: one 16-wide x-strip at fixed (z,y); sliding-window reuse
    const int xs = (tid & 7) << 4;      // x0 in {0,16,...,112}
    const int yy = (tid >> 3) & 7;      // 0..7
    const int zz = tid >> 6;            // 0..3

    float acc[16];
    #pragma unroll
    for (int i = 0; i < 16; ++i) acc[i] = 0.0f;

    #pragma unroll
    for (int kz = 0; kz < 3; ++kz) {
        #pragma unroll
        for (int ky = 0; ky < 3; ++ky) {
            const float* rp = &lds[((zz + kz) * LY + (yy + ky)) * LROW + 3 + xs];
            float r[18];
            #pragma unroll
            for (int i = 0; i < 18; ++i) r[i] = rp[i];
            #pragma unroll
            for (int kx = 0; kx < 3; ++kx) {
                const float wv = wk[(kz * 3 + ky) * 3 + kx];
                #pragma unroll
                for (int i = 0; i < 16; ++i)
                    acc[i] = fmaf(r[i + kx], wv, acc[i]);
            }
        }
    }

    // 5) partial statistics + store conv result (y lives in d_out for pass 2)
    float s1 = 0.0f, s2 = 0.0f;
    #pragma unroll
    for (int i = 0; i < 16; ++i) { s1 += acc[i]; s2 += acc[i] * acc[i]; }

    const size_t obase = base + ((size_t)(d0 + zz) * H_ + (h0 + yy)) * W_ + xs;
    #pragma unroll
    for (int j = 0; j < 4; ++j) {
        float4 v = make_float4(acc[4*j], acc[4*j+1], acc[4*j+2], acc[4*j+3]);
        *(float4*)(y + obase + 4 * j) = v;
    }

    // 6) block tree-reduction in (reused) LDS, then one atomicAdd per block
    __syncthreads();                    // all LDS reads done; safe to reuse
    lds[tid] = s1;
    lds[256 + tid] = s2;
    __syncthreads();
    for (int off = 128; off > 0; off >>= 1) {
        if (tid < off) {
            lds[tid] += lds[tid + off];
            lds[256 + tid] += lds[256 + tid + off];
        }
        __syncthreads();
    }
    if (tid == 0) {
        const int bc = b * C_ + c;
        atomicAdd(&ws_sum[bc], lds[0]);
        atomicAdd(&ws_sumsq[bc], lds[256]);
    }
}

// ---------------------------------------------------------------------------
// Kernel 2: in-place InstanceNorm + ReLU over y (d_out). Grid (256, C, B).
// Each block handles 4096 contiguous floats of one (b,c) slice.
// ---------------------------------------------------------------------------
__global__ __launch_bounds__(256) void dwcb_norm_kernel(
    float* __restrict__ y, const float* __restrict__ ws_sum,
    const float* __restrict__ ws_sumsq,
    const float* __restrict__ gamma, const float* __restrict__ beta)
{
    const int c = blockIdx.y, b = blockIdx.z;
    const int bc = b * C_ + c;
    const float inv = 1.0f / (float)S_;
    const float mean = ws_sum[bc] * inv;
    const float var  = ws_sumsq[bc] * inv - mean * mean;
    const float rstd = rsqrtf(var + 1e-5f);
    const float sc = gamma[c] * rstd;
    const float sh = beta[c] - mean * sc;

    const size_t base = (size_t)bc * (size_t)S_ + (size_t)blockIdx.x * 4096;
    float4* p = (float4*)(y + base);
    #pragma unroll
    for (int j = 0; j < 4; ++j) {
        const int vi = threadIdx.x + 256 * j;    // coalesced float4 access
        float4 v = p[vi];
        v.x = fmaxf(fmaf(v.x, sc, sh), 0.0f);
        v.y = fmaxf(fmaf(v.y, sc, sh), 0.0f);
        v.z = fmaxf(fmaf(v.z, sc, sh), 0.0f);
        v.w = fmaxf(fmaf(v.w, sc, sh), 0.0f);
        p[vi] = v;
    }
}

// ---------------------------------------------------------------------------
extern "C" void kernel_launch(void* const* d_in, const int* in_sizes, int n_in,
                              void* d_out, int out_size, void* d_ws, size_t ws_size,
                              hipStream_t stream) {
    const float* x     = (const float*)d_in[0];
    const float* w     = (const float*)d_in[1];
    const float* gamma = (const float*)d_in[2];
    const float* beta  = (const float*)d_in[3];
    float* y  = (float*)d_out;
    float* ws = (float*)d_ws;              // [0..127] sum, [128..255] sumsq
    float* sum   = ws;
    float* sumsq = ws + B_ * C_;

    hipLaunchKernelGGL(dwcb_init_kernel, dim3(1), dim3(256), 0, stream, ws);

    dim3 grid(256, C_, B_);                // 16 d-tiles * 16 h-tiles per slice
    hipLaunchKernelGGL(dwcb_conv_stats_kernel, grid, dim3(256), 0, stream,
                       x, w, y, sum, sumsq);
    hipLaunchKernelGGL(dwcb_norm_kernel, grid, dim3(256), 0, stream,
                       y, sum, sumsq, gamma, beta);
}